// Sinkhorn_84370337563295
// MI455X (gfx1250) — compile-verified
//
#include <hip/hip_runtime.h>
#include <cmath>

// ---------------------------------------------------------------------------
// Sinkhorn (log-domain, Adam) for MI455X / gfx1250.
//
// K = log_alpha/tau is loop-invariant, so each logsumexp factorizes into a
// stabilized f32 GEMV over a precomputed matrix that stays resident in the
// 192 MB L2 (EtP + FP = 64 MB):
//   lse_row_i = Mrow_i + Vmax + log( sum_j E_ij * w_j ),  E_ij = exp(K_ij-Mrow_i)
// GEMV runs on V_WMMA_F32_16X16X4_F32 with the weight vector broadcast in A
// (all 16 A-rows identical -> every D row is the dot product) and the matrix
// tile in B, so 100% of the bandwidth-limiting B traffic is useful work.
//
// E/F are stored TILE-PACKED in exact WMMA B-register order:
//   addr = b*1M + nt*16384 + kt*64 + lane*2 + elem
//   (nt = n/16, kt = k/4, lane = ((k&2)>>1)*16 + (n&15), elem = k&1)
// so the inner loop is one aligned global_load_b64 per lane with immediate
// offsets (wave streams 8 KB contiguous), plus one ds_load_2addr_b64 for A.
//
// Final plan = exp(K + u + v) computed directly from the input (L2-resident).
// Workspace requirement: 2*8M + 8*8K floats ~= 64.3 MB.
// ---------------------------------------------------------------------------

#define BATCH 8
#define NN 1024
#define MM 1024
#define TAU_INV 20.0f
#define LOG_MARG -6.93147180559945f   /* -log(1024) */
#define AB1 0.9f
#define AB2 0.999f
#define EPS_ADAM 1e-8f
#define ADAM_LR 1.0f

typedef float v2f __attribute__((ext_vector_type(2)));
typedef float v8f __attribute__((ext_vector_type(8)));

// ---- Mrow[b,i] = max_j K[b,i,j] ------------------------------------------
__global__ void rowmax_kernel(const float* __restrict__ la, float* __restrict__ Mrow) {
  __shared__ float red[256];
  int row = blockIdx.x;                       // b*NN + i
  const float* p = la + (size_t)row * MM;
  int tid = threadIdx.x;
  float m = -INFINITY;
  for (int j = tid; j < MM; j += 256) m = fmaxf(m, p[j]);
  red[tid] = m; __syncthreads();
  for (int s = 128; s > 0; s >>= 1) {
    if (tid < s) red[tid] = fmaxf(red[tid], red[tid + s]);
    __syncthreads();
  }
  if (tid == 0) Mrow[row] = red[0] * TAU_INV;
}

// ---- Mcol[b,j] = max_i K[b,i,j] (coalesced column sweep) -----------------
__global__ void colmax_kernel(const float* __restrict__ la, float* __restrict__ Mcol) {
  int b = blockIdx.y;
  int j = blockIdx.x * 256 + threadIdx.x;
  const float* p = la + (size_t)b * NN * MM + j;
  float m = -INFINITY;
  for (int i = 0; i < NN; ++i) m = fmaxf(m, p[(size_t)i * MM]);
  Mcol[b * MM + j] = m * TAU_INV;
}

// ---- Build FP (col-pass matrix) and EtP (row-pass matrix), tile-packed ---
// F_ij = exp(K - Mcol_j): reduction k=i, output n=j.
// E_ij = exp(K - Mrow_i): reduction k=j, output n=i (i.e. E^T, packed).
// A 16x16 (i,j) input tile maps to one contiguous 256-float packed block:
//   FP  block base: jt*16384 + it*256      EtP block base: it*16384 + jt*256
__global__ void ef_kernel(const float* __restrict__ la, const float* __restrict__ Mrow,
                          const float* __restrict__ Mcol,
                          float* __restrict__ FP, float* __restrict__ EtP) {
  __shared__ float lf[256];
  __shared__ float le[256];
  int b = blockIdx.z, it = blockIdx.y, jt = blockIdx.x;
  int tx = threadIdx.x, ty = threadIdx.y;     // tx = j-in-tile, ty = i-in-tile
  int i = it * 16 + ty;
  int j = jt * 16 + tx;
  size_t base = (size_t)b * NN * MM;
  float K = la[base + (size_t)i * MM + j] * TAU_INV;
  float fval = __expf(K - Mcol[b * MM + j]);
  float eval = __expf(K - Mrow[b * NN + i]);
  // packed position within 256-float block: q*64 + lane*2 + elem
  int pos_f = ((ty >> 2) << 6) + ((((ty >> 1) & 1) << 4) + tx) * 2 + (ty & 1); // k=i,n=j
  int pos_e = ((tx >> 2) << 6) + ((((tx >> 1) & 1) << 4) + ty) * 2 + (tx & 1); // k=j,n=i
  lf[pos_f] = fval;
  le[pos_e] = eval;
  __syncthreads();
  int tid = ty * 16 + tx;
  FP [base + (size_t)jt * 16384 + (size_t)it * 256 + tid] = lf[tid];
  EtP[base + (size_t)it * 16384 + (size_t)jt * 256 + tid] = le[tid];
}

__global__ void zero_kernel(float* __restrict__ p, int n) {
  int i = blockIdx.x * 256 + threadIdx.x;
  if (i < n) p[i] = 0.0f;
}

// ---- One Sinkhorn half-step (shared by row pass and col pass) ------------
// Mat: tile-packed [b][nt][kt][lane][2].  16 outputs (one nt) per block,
// 8 waves each own 32 consecutive kt tiles (128 of the 1024-long k dim).
__global__ void half_step_kernel(const float* __restrict__ Mat,
                                 const float* __restrict__ margMax,
                                 const float* __restrict__ other,
                                 float* __restrict__ s, float* __restrict__ ms,
                                 float* __restrict__ vs,
                                 float bc1, float bc2) {
  __shared__ float w[1024];
  __shared__ float red[256];
  __shared__ float partS[8][16];
  int b  = blockIdx.x >> 6;
  int nt = blockIdx.x & 63;                   // output tile (16 outputs)
  int tid = threadIdx.x;

  // --- max of dual potential + stabilized exp weights into LDS ---
  const float* ob = other + b * 1024;
  float lm = -INFINITY;
  for (int k = tid; k < 1024; k += 256) lm = fmaxf(lm, ob[k]);
  red[tid] = lm; __syncthreads();
  for (int st = 128; st > 0; st >>= 1) {
    if (tid < st) red[tid] = fmaxf(red[tid], red[tid + st]);
    __syncthreads();
  }
  float omax = red[0];
  for (int k = tid; k < 1024; k += 256) w[k] = __expf(ob[k] - omax);
  __syncthreads();

  // --- WMMA GEMV over packed B tiles ---
  // f32 16x16x4 A layout: lanes 0-15 hold k=0(v0),k=1(v1); lanes 16-31 k=2,k=3.
  int wave = tid >> 5, lane = tid & 31;
  int kb = (lane >> 4) << 1;                  // 0 or 2
  int k0 = wave << 7;                         // wave's k-range base
  const float* p = Mat + (size_t)b * NN * MM + ((size_t)nt << 14)
                 + ((size_t)wave << 11) + (lane << 1);
  v8f acc = {0.f, 0.f, 0.f, 0.f, 0.f, 0.f, 0.f, 0.f};
  #pragma unroll 8
  for (int kt = 0; kt < 32; ++kt) {
    v2f Bv = *(const v2f*)(p + (kt << 6));    // one aligned b64, imm offset
    int krow = k0 + (kt << 2) + kb;
    v2f A; A[0] = w[krow]; A[1] = w[krow + 1];
    acc = __builtin_amdgcn_wmma_f32_16x16x4_f32(
        /*neg_a=*/false, A, /*neg_b=*/false, Bv,
        /*c_mod=*/(short)0, acc, /*reuse_a=*/false, /*reuse_b=*/false);
  }
  // D rows are all identical (A rows identical): VGPR0, lanes 0-15 = S[n].
  if (lane < 16) partS[wave][lane] = acc[0];
  __syncthreads();

  // --- combine wave partials, logsumexp, Adam update for 16 outputs ---
  if (tid < 16) {
    float S = 0.f;
    #pragma unroll
    for (int wv = 0; wv < 8; ++wv) S += partS[wv][tid];
    int gi = b * 1024 + (nt << 4) + tid;
    float lse = margMax[gi] + omax + __logf(S);
    float g  = (LOG_MARG - lse) - s[gi];
    float m_ = AB1 * ms[gi] + (1.0f - AB1) * g;
    float v_ = AB2 * vs[gi] + (1.0f - AB2) * g * g;
    ms[gi] = m_; vs[gi] = v_;
    s[gi] += ADAM_LR * (m_ / bc1) / (sqrtf(v_ / bc2) + EPS_ADAM);
  }
}

// ---- plan = exp(K + u_i + v_j), straight from the (L2-resident) input ----
__global__ void plan_kernel(const float* __restrict__ la, const float* __restrict__ u,
                            const float* __restrict__ v, float* __restrict__ out) {
  size_t q = (size_t)blockIdx.x * 256 + threadIdx.x;   // quad index
  size_t idx = q * 4;
  int b = (int)(idx >> 20);
  int i = (int)((idx >> 10) & 1023);
  int j = (int)(idx & 1023);
  float4 k4 = *(const float4*)(la + idx);
  float4 v4 = *(const float4*)(v + b * 1024 + j);
  float  uu = u[b * 1024 + i];
  float4 o;
  o.x = __expf(k4.x * TAU_INV + uu + v4.x);
  o.y = __expf(k4.y * TAU_INV + uu + v4.y);
  o.z = __expf(k4.z * TAU_INV + uu + v4.z);
  o.w = __expf(k4.w * TAU_INV + uu + v4.w);
  *(float4*)(out + idx) = o;
}

extern "C" void kernel_launch(void* const* d_in, const int* in_sizes, int n_in,
                              void* d_out, int out_size, void* d_ws, size_t ws_size,
                              hipStream_t stream) {
  (void)in_sizes; (void)n_in; (void)out_size; (void)ws_size;
  const float* la = (const float*)d_in[0];
  float* out = (float*)d_out;
  float* ws  = (float*)d_ws;

  const size_t MAT = (size_t)BATCH * NN * MM;   // 8M floats
  const int    VEC = BATCH * NN;                // 8192

  float* FP   = ws;                 // col-pass matrix, tile-packed
  float* EtP  = ws + MAT;           // row-pass matrix (E^T), tile-packed
  float* Mrow = ws + 2 * MAT;
  float* Mcol = Mrow + VEC;
  float* u    = Mcol + VEC;
  float* mu   = u  + VEC;
  float* vu   = mu + VEC;
  float* v    = vu + VEC;
  float* mv   = v  + VEC;
  float* vv   = mv + VEC;

  // One-time precompute (only HBM-heavy phase; iteration matrices then live in L2).
  rowmax_kernel<<<BATCH * NN, 256, 0, stream>>>(la, Mrow);
  colmax_kernel<<<dim3(MM / 256, BATCH), 256, 0, stream>>>(la, Mcol);
  ef_kernel<<<dim3(MM / 16, NN / 16, BATCH), dim3(16, 16), 0, stream>>>(la, Mrow, Mcol, FP, EtP);
  zero_kernel<<<(6 * VEC) / 256, 256, 0, stream>>>(u, 6 * VEC);  // u,mu,vu,v,mv,vv contiguous

  for (int t = 1; t <= 10; ++t) {
    float bc1 = 1.0f - powf(AB1, (float)t);
    float bc2 = 1.0f - powf(AB2, (float)t);
    // u half-step: reduce over j using EtP, weights from v
    half_step_kernel<<<BATCH * (NN / 16), 256, 0, stream>>>(EtP, Mrow, v, u, mu, vu, bc1, bc2);
    // v half-step: reduce over i using FP, weights from freshly-updated u
    half_step_kernel<<<BATCH * (MM / 16), 256, 0, stream>>>(FP, Mcol, u, v, mv, vv, bc1, bc2);
  }

  plan_kernel<<<(unsigned)((MAT / 4) / 256), 256, 0, stream>>>(la, u, v, out);
}